// DynamicSemanticMemory_70205535420794
// MI455X (gfx1250) — compile-verified
//
#include <hip/hip_runtime.h>
#include <math.h>

typedef float v2f __attribute__((ext_vector_type(2)));
typedef float v8f __attribute__((ext_vector_type(8)));

static constexpr int Dm = 384;   // embedding dim
static constexpr int Bn = 2048;  // batch / steps
static constexpr int Cn = 100;   // max centroids

// scal indices
#define S_N2   0
#define S_COEF 1
#define S_SSC  2
#define S_MOM  3
#define S_VINV 4
// iscal indices
#define I_ACTIVE 0
#define I_WIDX   1
#define I_DOW    2

__device__ __forceinline__ v8f wmma4(v2f a, v2f b, v8f c) {
  // D = A(16x4 f32) * B(4x16 f32) + C  -> v_wmma_f32_16x16x4_f32
  return __builtin_amdgcn_wmma_f32_16x16x4_f32(false, a, false, b, (short)0, c,
                                               false, false);
}

__device__ __forceinline__ float wave_sum(float v) {
#pragma unroll
  for (int o = 16; o > 0; o >>= 1) v += __shfl_down(v, o, 32);
  return v;
}

// ---- CDNA5 async global->LDS copy (ASYNCcnt-tracked DMA into LDS) ----------
__device__ __forceinline__ void async_g2l_b128(unsigned lds_addr,
                                               const void* gaddr) {
  // copies 16 bytes from global address gaddr to LDS address lds_addr, tracked by ASYNCcnt
  asm volatile("global_load_async_to_lds_b128 %0, %1, off"
               :
               : "v"(lds_addr), "v"(gaddr)
               : "memory");
}
__device__ __forceinline__ void wait_async_le1() {
  asm volatile("s_wait_asynccnt 0x1" ::: "memory");
}
__device__ __forceinline__ void wait_async_0() {
  asm volatile("s_wait_asynccnt 0x0" ::: "memory");
}

// ---------------- Phase 1: Gram matrix G = E * E^T (2048x2048) --------------
// 128x128 tiles of 16x16; one wave per tile; k-loop over 384 in steps of 4.
__global__ __launch_bounds__(256) void gram_kernel(const float* __restrict__ E,
                                                   float* __restrict__ G) {
  int wv = (blockIdx.x << 3) | (threadIdx.x >> 5);  // global wave id, 16384 total
  int lane = threadIdx.x & 31;
  int ti = wv >> 7, tj = wv & 127;
  int m = lane & 15, h = lane >> 4;
  const float* ra = E + (size_t)(ti * 16 + m) * Dm;
  const float* rb = E + (size_t)(tj * 16 + m) * Dm;
  v8f c = {};
  for (int k = 0; k < Dm; k += 4) {
    v2f a, b;
    // A 16x4: VGPR0 -> K=0(l0-15)/K=2(l16-31); VGPR1 -> K=1 / K=3
    a.x = ra[k + 2 * h];
    a.y = ra[k + 2 * h + 1];
    // B 4x16 (rows of E_j as columns of E^T): VGPR0 -> K=h; VGPR1 -> K=h+2
    b.x = rb[k + h];
    b.y = rb[k + h + 2];
    c = wmma4(a, b, c);
  }
  float* o = G + (size_t)(ti * 16) * Bn + tj * 16;
#pragma unroll
  for (int r = 0; r < 8; ++r)
    o[(size_t)(r + 8 * h) * Bn + m] = c[r];  // C: VGPR r -> M=r / M=r+8
}

// ---------- Phase 2: qinit[j] = (d/||e_j||^2) * e_j^T K0 e_j ---------------
// Block handles 16 rows; 8 waves cover 24 column tiles (3 each) of M = E*K0.
__global__ __launch_bounds__(256) void quadform_kernel(
    const float* __restrict__ E, const float* __restrict__ K0,
    float* __restrict__ qinit) {
  __shared__ float zsh[16];
  int tid = threadIdx.x, wv = tid >> 5, lane = tid & 31;
  int m = lane & 15, h = lane >> 4;
  int j0 = blockIdx.x * 16;
  if (tid < 16) zsh[tid] = 0.f;
  __syncthreads();
  float acc[8];
#pragma unroll
  for (int r = 0; r < 8; ++r) acc[r] = 0.f;
  const float* ra = E + (size_t)(j0 + m) * Dm;
  for (int s = 0; s < 3; ++s) {
    int n0 = (wv * 3 + s) * 16;
    v8f c = {};
    for (int k = 0; k < Dm; k += 4) {
      v2f a, b;
      a.x = ra[k + 2 * h];
      a.y = ra[k + 2 * h + 1];
      b.x = K0[(size_t)(k + h) * Dm + n0 + m];
      b.y = K0[(size_t)(k + h + 2) * Dm + n0 + m];
      c = wmma4(a, b, c);
    }
#pragma unroll
    for (int r = 0; r < 8; ++r)
      acc[r] += c[r] * E[(size_t)(j0 + r + 8 * h) * Dm + n0 + m];
  }
#pragma unroll
  for (int r = 0; r < 8; ++r) atomicAdd(&zsh[r + 8 * h], acc[r]);
  __syncthreads();
  if (tid < 16) {
    const float* e = E + (size_t)(j0 + tid) * Dm;
    float g = 0.f;
    for (int k = 0; k < Dm; ++k) g += e[k] * e[k];
    float sA = (g > 1e-8f) ? (float)Dm / g : 1.f;
    qinit[j0 + tid] = zsh[tid] * sA;
  }
}

// ---------------- Phase 3: sequential scan (single WGP) ---------------------
// LDS: centroids + q vector + Gram-row double buffer (async DMA) + scalars.
static constexpr size_t SCAN_SMEM_BYTES =
    (size_t)(Cn * Dm + 5 * Bn + 2 * Bn /*gbuf*/ + 2 * Dm + 128 + 128 + 1024 +
             16) * sizeof(float) +
    (size_t)(128 + 16) * sizeof(int);

__global__ __launch_bounds__(1024) void scan_kernel(
    const float* __restrict__ E, const float* __restrict__ K0,
    const float* __restrict__ G, const float* __restrict__ qinit,
    float* __restrict__ warr, float* __restrict__ scalg,
    float* __restrict__ novOut, float* __restrict__ centOut,
    float* __restrict__ activeOut) {
  extern __shared__ float smem[];
  float* cent   = smem;            // Cn*Dm
  float* qv     = cent + Cn * Dm;  // Bn : q_j = <K, P_j>
  float* gdiag  = qv + Bn;         // Bn : ||e_j||^2
  float* sAL    = gdiag + Bn;      // Bn : trace scale d/||e||^2
  float* coefL  = sAL + Bn;        // Bn : c_t
  float* sL     = coefL + Bn;      // Bn : s_t (norm clamp)
  float* gbuf   = sL + Bn;         // 2*Bn : Gram-row double buffer
  float* eL     = gbuf + 2 * Bn;   // Dm
  float* vtmp   = eL + Dm;         // Dm
  float* simbuf = vtmp + Dm;       // 128
  float* cnorm  = simbuf + 128;    // 128
  float* red    = cnorm + 128;     // 1024
  float* scal   = red + 1024;      // 16
  int*   counts = (int*)(scal + 16);  // 128
  int*   iscal  = counts + 128;       // 16

  const int tid = threadIdx.x;
  const int wv = tid >> 5, lane = tid & 31;
  // LDS byte offset of gbuf (flat LDS-aperture addresses carry the LDS byte
  // offset in addr[31:0] on CDNA5).
  const unsigned gbase = (unsigned)(uintptr_t)gbuf;

  // ---- init ----
  for (int i = tid; i < Cn * Dm; i += 1024) cent[i] = 0.f;
  if (tid < 128) { cnorm[tid] = 0.f; counts[tid] = 0; }
  for (int j = tid; j < Bn; j += 1024) {
    qv[j] = qinit[j];
    float g = G[(size_t)j * Bn + j];
    gdiag[j] = g;
    sAL[j] = (g > 1e-8f) ? (float)Dm / g : 1.f;
  }
  {
    float p = 0.f;
    for (int i = tid; i < Dm * Dm; i += 1024) { float v = K0[i]; p += v * v; }
    red[tid] = p;
  }
  if (tid == 0) iscal[I_ACTIVE] = 0;
  // kick off async DMA of Gram row 0 into gbuf[0] (512 lanes x 16B = 8KB)
  if (tid < 512) async_g2l_b128(gbase + tid * 16u, G + tid * 4);
  __syncthreads();
  for (int s = 512; s > 0; s >>= 1) {
    if (tid < s) red[tid] += red[tid + s];
    __syncthreads();
  }
  if (tid == 0) scal[S_N2] = red[0];  // ||K0||_F^2
  __syncthreads();

  for (int t = 0; t < Bn; ++t) {
    // issue async DMA for row t+1 into the other half of the double buffer,
    // then wait for row t's DMA to land (ordered completion => <=1 is enough).
    if (tid < 512) {
      if (t + 1 < Bn) {
        unsigned dst = gbase + (unsigned)((((t + 1) & 1) * Bn + tid * 4) * 4);
        async_g2l_b128(dst, G + (size_t)(t + 1) * Bn + tid * 4);
        wait_async_le1();
      } else {
        wait_async_0();
      }
    }
    if (tid < Dm) eL[tid] = E[(size_t)t * Dm + tid];
    __syncthreads();
    const float g_tt = gdiag[t];
    const float enorm = sqrtf(fmaxf(g_tt, 0.f));
    const float invE = 1.f / fmaxf(enorm, 1e-12f);

    // sims: each wave handles centroids c = wv, wv+32, wv+64, wv+96
    for (int c = wv; c < Cn; c += 32) {
      float p = 0.f;
      const float* cc = cent + c * Dm;
      for (int k = lane; k < Dm; k += 32) p += cc[k] * eL[k];
      p = wave_sum(p);
      if (lane == 0) {
        float s = p * invE / fmaxf(cnorm[c], 1e-12f);
        simbuf[c] = fminf(fmaxf(s, 0.f), 1.f);
      }
    }
    __syncthreads();

    // wave 0: argmax + scalar recurrence
    if (tid < 32) {
      int active = iscal[I_ACTIVE];
      float bv = -INFINITY;
      int bi = 0x7fffffff;
      for (int c = lane; c < Cn; c += 32) {
        float v = (c < active) ? simbuf[c] : -INFINITY;
        if (v > bv) { bv = v; bi = c; }
      }
#pragma unroll
      for (int o = 16; o > 0; o >>= 1) {
        float ov = __shfl_down(bv, o, 32);
        int oi = __shfl_down(bi, o, 32);
        if (ov > bv || (ov == bv && oi < bi)) { bv = ov; bi = oi; }
      }
      if (lane == 0) {
        int cluster; float nov;
        if (active > 0) {
          cluster = bi;
          float x = fminf(fmaxf(1.f - bv, 0.f), 1.f);
          nov = fminf(fmaxf(sqrtf(x), 0.f), 1.f);
        } else { cluster = -1; nov = 1.f; }

        float N2 = scal[S_N2];
        float curn = sqrtf(fmaxf(N2, 0.f));
        float cap = curn / 50.f;
        float lr = (cap > 0.96f) ? 0.f : 1e-6f * expf(cap * -13.8155106f); // 3*ln(.01)
        float sAt = sAL[t];
        float pn = sAt * g_tt;   // ||P_t||_F == d when trace-normalized
        float cP = lr * nov;
        float un = cP * pn;
        if (un > 0.1f) cP *= 0.1f / un;
        float qt = qv[t];
        float N2n = N2 + 2.f * cP * qt + cP * cP * pn * pn;
        float kn = sqrtf(fmaxf(N2n, 0.f));
        float sc = (kn > 50.f) ? 50.f / kn : 1.f;
        scal[S_N2] = N2n * sc * sc;
        scal[S_COEF] = cP;
        scal[S_SSC] = sc;
        coefL[t] = cP;
        sL[t] = sc;
        novOut[t] = nov;

        bool grow = (nov > 0.8f) && (active < Cn);
        bool upd = (!grow) && (cluster >= 0);
        int widx = grow ? active : (cluster > 0 ? cluster : 0);
        int cnt = counts[widx];
        scal[S_MOM] = fminf(0.99f, (float)cnt / ((float)cnt + 1.f));
        iscal[I_WIDX] = widx;
        iscal[I_DOW] = grow ? 1 : (upd ? 2 : 0);
        if (grow) { counts[widx] = 1; iscal[I_ACTIVE] = active + 1; }
        else if (upd) counts[widx] = cnt + 1;
      }
    }
    __syncthreads();

    // q_j <- s_t * (q_j + c_t * <P_t, P_j>)  using LDS-resident Gram row t
    {
      const float cP = scal[S_COEF], sc = scal[S_SSC], sAt = sAL[t];
      const float* gr = gbuf + (t & 1) * Bn;
      for (int j = tid; j < Bn; j += 1024) {
        float g = gr[j];
        float g2 = sAt * sAL[j] * g * g;
        qv[j] = sc * (qv[j] + cP * g2);
      }
    }
    // centroid candidate value
    {
      int dow = iscal[I_DOW];
      if (dow != 0 && tid < Dm) {
        int w = iscal[I_WIDX];
        float mom = scal[S_MOM];
        vtmp[tid] = (dow == 1) ? eL[tid]
                               : mom * cent[w * Dm + tid] + (1.f - mom) * eL[tid];
      }
    }
    __syncthreads();
    if (tid < 32) {
      int dow = iscal[I_DOW];
      if (dow == 2) {
        float p = 0.f;
        for (int k = lane; k < Dm; k += 32) p += vtmp[k] * vtmp[k];
        p = wave_sum(p);
        if (lane == 0) {
          float vn = sqrtf(p);
          float inv = 1.f / fmaxf(vn, 1e-12f);
          scal[S_VINV] = inv;
          cnorm[iscal[I_WIDX]] = vn * inv;
        }
      } else if (dow == 1 && lane == 0) {
        cnorm[iscal[I_WIDX]] = enorm;  // grow stores raw emb
      }
    }
    __syncthreads();
    {
      int dow = iscal[I_DOW];
      if (dow != 0 && tid < Dm) {
        int w = iscal[I_WIDX];
        cent[w * Dm + tid] = (dow == 1) ? vtmp[tid] : vtmp[tid] * scal[S_VINV];
      }
    }
    __syncthreads();
  }

  // ---- finalize: suffix products -> rank-1 weights ----
  if (tid == 0) {
    float suf = 1.f;
    for (int t = Bn - 1; t >= 0; --t) {
      suf *= sL[t];
      warr[t] = coefL[t] * suf * sAL[t];  // weight on e_t e_t^T
    }
    scalg[0] = suf;  // product of all s -> scale on K0
    *activeOut = (float)iscal[I_ACTIVE];
  }
  for (int i = tid; i < Cn * Dm; i += 1024) centOut[i] = cent[i];
}

// -------- Phase 4: K = pS*K0 + E^T diag(w) E   (384x384, k=2048, WMMA) ------
__global__ __launch_bounds__(256) void finalk_kernel(
    const float* __restrict__ E, const float* __restrict__ K0,
    const float* __restrict__ warr, const float* __restrict__ scalg,
    float* __restrict__ out) {
  int wvg = (blockIdx.x << 3) | (threadIdx.x >> 5);  // 576 tiles = 24x24
  int lane = threadIdx.x & 31;
  int ta = wvg / 24, tb = wvg % 24;
  int m = lane & 15, h = lane >> 4;
  int a0 = ta * 16 + m, b0 = tb * 16 + m;
  v8f c = {};
  for (int k = 0; k < Bn; k += 4) {
    v2f a, b;
    a.x = E[(size_t)(k + 2 * h) * Dm + a0] * warr[k + 2 * h];
    a.y = E[(size_t)(k + 2 * h + 1) * Dm + a0] * warr[k + 2 * h + 1];
    b.x = E[(size_t)(k + h) * Dm + b0];
    b.y = E[(size_t)(k + h + 2) * Dm + b0];
    c = wmma4(a, b, c);
  }
  float pS = scalg[0];
#pragma unroll
  for (int r = 0; r < 8; ++r) {
    int aa = ta * 16 + r + 8 * h;
    int bb = tb * 16 + m;
    out[(size_t)aa * Dm + bb] = pS * K0[(size_t)aa * Dm + bb] + c[r];
  }
}

extern "C" void kernel_launch(void* const* d_in, const int* in_sizes, int n_in,
                              void* d_out, int out_size, void* d_ws,
                              size_t ws_size, hipStream_t stream) {
  (void)in_sizes; (void)n_in; (void)out_size; (void)ws_size;
  const float* E  = (const float*)d_in[0];   // [2048,384]
  const float* K0 = (const float*)d_in[1];   // [384,384]
  // d_in[2] (centroids zeros) / d_in[3] (counts zeros) re-created in-kernel.

  float* ws    = (float*)d_ws;
  float* G     = ws;                          // 2048*2048
  float* qinit = G + (size_t)Bn * Bn;         // 2048
  float* warr  = qinit + Bn;                  // 2048
  float* scalg = warr + Bn;                   // few scalars

  float* out       = (float*)d_out;
  float* kernelOut = out;                     // 384*384
  float* novOut    = out + Dm * Dm;           // 2048
  float* centOut   = novOut + Bn;             // 100*384
  float* activeOut = centOut + Cn * Dm;       // 1

  gram_kernel<<<dim3(2048), dim3(256), 0, stream>>>(E, G);
  quadform_kernel<<<dim3(128), dim3(256), 0, stream>>>(E, K0, qinit);
  hipFuncSetAttribute((const void*)scan_kernel,
                      hipFuncAttributeMaxDynamicSharedMemorySize,
                      (int)SCAN_SMEM_BYTES);
  scan_kernel<<<dim3(1), dim3(1024), SCAN_SMEM_BYTES, stream>>>(
      E, K0, G, qinit, warr, scalg, novOut, centOut, activeOut);
  finalk_kernel<<<dim3(72), dim3(256), 0, stream>>>(E, K0, warr, scalg,
                                                    kernelOut);
}